// TransD_76020921140302
// MI455X (gfx1250) — compile-verified
//
#include <hip/hip_runtime.h>
#include <cstdint>

// TransD projection: memory-bound gather + dot + axpy.
//   out[0 : B*D)        = h + p_r * <p_h, h>
//   out[B*D : 2B*D)     = r                       (pure copy, via async LDS path)
//   out[2B*D : 3B*D)    = t + p_r * <p_t, t>
// B = 65536, D = 256, all fp32. ~576 MB of traffic -> HBM-bound (~25 us floor
// at 23.3 TB/s); WMMA deliberately unused (diag-of-matmul wastes 15/16 MACs
// and saves zero bytes). CDNA5-specific path used: async global<->LDS DMA.

#define DIM_D 256
#define BATCH_N 65536
#define WAVES_PER_BLOCK 8

typedef float f4 __attribute__((ext_vector_type(4)));

__device__ __forceinline__ float dot4(f4 a, f4 b) {
  return a.x * b.x + a.y * b.y + a.z * b.z + a.w * b.w;
}

// --- CDNA5 async global<->LDS (ASYNCcnt-tracked), per CDNA5 ISA ---
__device__ __forceinline__ void async_ld_b128(uint32_t lds_off, const void* gptr) {
  asm volatile("global_load_async_to_lds_b128 %0, %1, off"
               :: "v"(lds_off), "v"(gptr) : "memory");
}
__device__ __forceinline__ void async_st_b128(void* gptr, uint32_t lds_off) {
  asm volatile("global_store_async_from_lds_b128 %0, %1, off"
               :: "v"(gptr), "v"(lds_off) : "memory");
}
__device__ __forceinline__ void wait_async0() {
  asm volatile("s_wait_asynccnt 0" ::: "memory");
}

__global__ __launch_bounds__(256) void transd_kernel(
    const int* __restrict__ trip,        // (B, 3) int32: h, r, t
    const float* __restrict__ ent_embs,  // (N_ENT+1, D)
    const float* __restrict__ ent_maps,  // (N_ENT+1, D)
    const float* __restrict__ rel_embs,  // (N_REL+1, D)
    const float* __restrict__ rel_maps,  // (N_REL+1, D)
    float* __restrict__ out)             // (3*B*D)
{
  __shared__ float rbuf[WAVES_PER_BLOCK * DIM_D];  // 8 KB staging for r-row copies

  const int wave = threadIdx.x >> 5;
  const int lane = threadIdx.x & 31;
  const int elem = blockIdx.x * WAVES_PER_BLOCK + wave;

  // Wave-uniform indices -> force scalar so row bases use SADDR-form loads.
  int h_idx = trip[3 * elem + 0];
  int r_idx = trip[3 * elem + 1];
  int t_idx = trip[3 * elem + 2];
  h_idx = __builtin_amdgcn_readfirstlane(h_idx);
  r_idx = __builtin_amdgcn_readfirstlane(r_idx);
  t_idx = __builtin_amdgcn_readfirstlane(t_idx);

  // Kick off r-row copy (global -> LDS) early; it overlaps the VALU work below
  // and never touches VGPRs for data.
  const uint32_t lds0 =
      (uint32_t)(uintptr_t)(&rbuf[wave * DIM_D]) + (uint32_t)lane * 16u;
  const float* rsrc = rel_embs + (size_t)r_idx * DIM_D + lane * 4;
  async_ld_b128(lds0, rsrc);                 // floats [0,128)
  async_ld_b128(lds0 + 512u, rsrc + 128);    // floats [128,256)

  // Each lane owns 8 floats of each row as two b128 loads
  // (512 B contiguous per instruction across the wave -> fully coalesced).
  const f4* he = (const f4*)(ent_embs + (size_t)h_idx * DIM_D);
  const f4* hm = (const f4*)(ent_maps + (size_t)h_idx * DIM_D);
  const f4* te = (const f4*)(ent_embs + (size_t)t_idx * DIM_D);
  const f4* tm = (const f4*)(ent_maps + (size_t)t_idx * DIM_D);
  const f4* pr = (const f4*)(rel_maps + (size_t)r_idx * DIM_D);

  // Entity tables: 512 MB, ~no reuse -> non-temporal (don't thrash L2).
  f4 h0  = __builtin_nontemporal_load(he + lane);
  f4 h1  = __builtin_nontemporal_load(he + 32 + lane);
  f4 ph0 = __builtin_nontemporal_load(hm + lane);
  f4 ph1 = __builtin_nontemporal_load(hm + 32 + lane);
  f4 t0  = __builtin_nontemporal_load(te + lane);
  f4 t1  = __builtin_nontemporal_load(te + 32 + lane);
  f4 pt0 = __builtin_nontemporal_load(tm + lane);
  f4 pt1 = __builtin_nontemporal_load(tm + 32 + lane);
  // Relation map table: 1 MB, hot -> normal (L2-resident) loads.
  f4 pr0 = pr[lane];
  f4 pr1 = pr[32 + lane];

  // 256-wide dot products: 8 lane-local MACs + 5-step wave32 xor reduction.
  float dh = dot4(ph0, h0) + dot4(ph1, h1);
  float dt = dot4(pt0, t0) + dot4(pt1, t1);
#pragma unroll
  for (int m = 16; m >= 1; m >>= 1) {
    dh += __shfl_xor(dh, m, 32);
    dt += __shfl_xor(dt, m, 32);
  }

  f4 oh0 = h0 + pr0 * dh;
  f4 oh1 = h1 + pr1 * dh;
  f4 ot0 = t0 + pr0 * dt;
  f4 ot1 = t1 + pr1 * dt;

  // Outputs are streamed (192 MB, never re-read) -> non-temporal stores.
  f4* outh = (f4*)(out + (size_t)elem * DIM_D);
  f4* outt = (f4*)(out + (size_t)(2 * BATCH_N) * DIM_D + (size_t)elem * DIM_D);
  __builtin_nontemporal_store(oh0, outh + lane);
  __builtin_nontemporal_store(oh1, outh + 32 + lane);
  __builtin_nontemporal_store(ot0, outt + lane);
  __builtin_nontemporal_store(ot1, outt + 32 + lane);

  // Finish r-row copy: LDS -> global (async, drains at implicit wait in endpgm).
  wait_async0();  // ensure both async loads landed in LDS
  float* rdst = out + (size_t)BATCH_N * DIM_D + (size_t)elem * DIM_D + lane * 4;
  async_st_b128(rdst, lds0);
  async_st_b128(rdst + 128, lds0 + 512u);
}

extern "C" void kernel_launch(void* const* d_in, const int* in_sizes, int n_in,
                              void* d_out, int out_size, void* d_ws, size_t ws_size,
                              hipStream_t stream) {
  const int*   trip     = (const int*)d_in[0];
  const float* ent_embs = (const float*)d_in[1];
  const float* ent_maps = (const float*)d_in[2];
  const float* rel_embs = (const float*)d_in[3];
  const float* rel_maps = (const float*)d_in[4];
  float* out = (float*)d_out;

  dim3 grid(BATCH_N / WAVES_PER_BLOCK);  // 8192 blocks x 256 threads (8 wave32/block)
  transd_kernel<<<grid, 256, 0, stream>>>(trip, ent_embs, ent_maps,
                                          rel_embs, rel_maps, out);
  (void)in_sizes; (void)n_in; (void)out_size; (void)d_ws; (void)ws_size;
}